// bicon_loss_71588514889850
// MI455X (gfx1250) — compile-verified
//
#include <hip/hip_runtime.h>
#include <cstdint>

// Problem constants (from reference): B=16, C=8, H=352, W=352
#define BB 16
#define CC 8
#define HH 352
#define WW 352
#define TH 8
#define TW 32
#define HALO_H (TH + 2)               // 10
#define HALO_W (TW + 2)               // 34
#define HALO_PER_C (HALO_H * HALO_W)  // 340
#define HALO_TOT (CC * HALO_PER_C)    // 2720 (divisible by 4)
#define TILES_X (WW / TW)             // 11
#define TILES_Y (HH / TH)             // 44
#define NBLK (BB * TILES_Y * TILES_X) // 7744

__device__ __forceinline__ float sigm(float x) {
  // v_exp_f32 + v_rcp_f32. sigm(-inf): exp(+inf)=inf -> rcp(inf) = 0 exactly.
  return __builtin_amdgcn_rcpf(1.0f + __expf(-x));
}

__device__ __forceinline__ float bce_term(float p, int t) {
  // reference: clip(p, 1e-12, 1-1e-12); t in {0,1}.
  // Single log per site: t ? log(p) : log1p(-p) == log(t ? p : 1-p) to ~1e-12
  // absolute (1-p exact for p>=0.5; p->1 still gives -inf like the reference).
  p = fminf(fmaxf(p, 1e-12f), 1.0f - 1e-12f);
  return __logf(t ? p : 1.0f - p);
}

// gfx1250 async global->LDS copy (ASYNCcnt-tracked). lds_off = LDS byte
// offset (low 32 bits of the flat pointer to __shared__).
__device__ __forceinline__ void async_cp(uint32_t lds_off, const float* g) {
  asm volatile("global_load_async_to_lds_b32 %0, %1, off"
               :
               : "v"(lds_off), "v"((unsigned long long)(uintptr_t)g)
               : "memory");
}

__global__ void __launch_bounds__(256)
bicon_main(const float* __restrict__ pred,
           const int* __restrict__ tgt,
           const int* __restrict__ con,
           float* __restrict__ partial) {
  __shared__ __align__(16) float sp[HALO_TOT]; // halo: raw 'out' -> sigmoid
  __shared__ float wred[8][6];                 // per-wave partial sums

  const int tid = threadIdx.x;
  const int tx  = tid & 31;        // wave32 lane
  const int ty  = tid >> 5;        // wave id == tile row
  const int b   = blockIdx.z;
  const int h0  = blockIdx.y * TH;
  const int w0  = blockIdx.x * TW;
  const size_t HW = (size_t)HH * WW;
  const float* pb = pred + (size_t)b * CC * HW;
  const float NEG_INF = -__builtin_inff();

  // ---- Row-based async staging: 80 halo rows (8 ch x 10 rows) of 34
  //      elements. Wave ty handles rows ty, ty+8, ... Lanes 0-31 copy
  //      x=0..31; lanes 0-1 copy x=32,33. Out-of-image cells are never
  //      loaded; they get -inf (sigmoid -> 0, matching pad-after-sigmoid).
#pragma unroll
  for (int k = 0; k < HALO_H; ++k) {
    const int rowId = ty + 8 * k;              // 0..79
    const int c = rowId / HALO_H;              // channel (div by 10: mul-shift)
    const int y = rowId - c * HALO_H;          // halo row within channel
    const int gh = h0 + y - 1;
    const bool rowValid = (unsigned)gh < (unsigned)HH;
    float* lrow = &sp[(c * HALO_H + y) * HALO_W];
    const float* grow =
        pb + (size_t)c * HW + (size_t)(rowValid ? gh : 0) * WW + (w0 - 1);
    {
      const int gw = w0 - 1 + tx;              // x = tx
      if (rowValid && (unsigned)gw < (unsigned)WW)
        async_cp((uint32_t)(uintptr_t)(lrow + tx), grow + tx);
      else
        lrow[tx] = NEG_INF;
    }
    if (tx < 2) {                              // x = 32, 33
      const int x = TW + tx;
      const int gw = w0 - 1 + x;
      if (rowValid && (unsigned)gw < (unsigned)WW)
        async_cp((uint32_t)(uintptr_t)(lrow + x), grow + x);
      else
        lrow[x] = NEG_INF;
    }
  }

  // ---- Overlap: per-pixel integer inputs while async copies fly. ----
  const int h = h0 + ty;
  const int w = w0 + tx;
  const int* cb = con + (size_t)b * CC * HW + (size_t)h * WW + w;
  int cv[8], csum = 0;
#pragma unroll
  for (int c = 0; c < 8; ++c) { cv[c] = cb[(size_t)c * HW]; csum += cv[c]; }
  const int tg = tgt[(size_t)b * HW + (size_t)h * WW + w];

  asm volatile("" ::: "memory");
  __builtin_amdgcn_s_wait_asynccnt(0);
  asm volatile("" ::: "memory");
  __syncthreads();

  // ---- In-place sigmoid over the halo (vectorized b128 LDS traffic):
  //      each element sigmoided ONCE (10.6/pixel); -inf sentinel -> 0. ----
  {
    float4* sp4 = (float4*)sp;
    for (int i = tid; i < HALO_TOT / 4; i += 256) {
      float4 t4 = sp4[i];
      t4.x = sigm(t4.x);
      t4.y = sigm(t4.y);
      t4.z = sigm(t4.z);
      t4.w = sigm(t4.w);
      sp4[i] = t4;
    }
  }
  __syncthreads();

  // ---- Per-pixel math: one thread per pixel of the 8x32 tile. ----
#define SIDX(c, y, x) (((c) * HALO_H + (y)) * HALO_W + (x))
  float pc[8], nb[8], v[8];
#pragma unroll
  for (int c = 0; c < 8; ++c) pc[c] = sp[SIDX(c, ty + 1, tx + 1)];
  // vote_out channel k = pc[k] * shifted-neighbor:
  nb[0] = sp[SIDX(7, ty,     tx    )];  // a7: c0 * c7[h-1,w-1]
  nb[1] = sp[SIDX(6, ty,     tx + 1)];  // a3: c1 * c6[h-1,w  ]
  nb[2] = sp[SIDX(5, ty,     tx + 2)];  // a5: c2 * c5[h-1,w+1]
  nb[3] = sp[SIDX(4, ty + 1, tx    )];  // a1: c3 * c4[h,  w-1]
  nb[4] = sp[SIDX(3, ty + 1, tx + 2)];  // a2: c4 * c3[h,  w+1]
  nb[5] = sp[SIDX(2, ty + 2, tx    )];  // a6: c5 * c2[h+1,w-1]
  nb[6] = sp[SIDX(1, ty + 2, tx + 1)];  // a4: c6 * c1[h+1,w  ]
  nb[7] = sp[SIDX(0, ty + 2, tx + 2)];  // a8: c7 * c0[h+1,w+1]

  float vsum = 0.0f, vmin = 3.0e38f;
#pragma unroll
  for (int k = 0; k < 8; ++k) {
    v[k] = pc[k] * nb[k];
    vsum += v[k];
    vmin = fminf(vmin, v[k]);
  }
  const float glo = 0.125f * vsum;  // glo_map

  const bool edge = (csum > 0) && (csum < 8);
  const float dm = edge ? (1.0f - vmin) : glo;  // decouple_map

  float a_bi = 0.0f, a_con = 0.0f;
#pragma unroll
  for (int c = 0; c < 8; ++c) {
    a_con += bce_term(pc[c], cv[c]);
    a_bi  += bce_term(v[c],  cv[c]);
  }

  // 6 running sums: [0]=decouple-bce, [1]=sum(glo), [2]=sum(tgt),
  // [3]=sum(glo*tgt), [4]=bimap-bce, [5]=conmap-bce
  float r6[6];
  r6[0] = bce_term(dm, tg);
  r6[1] = glo;
  r6[2] = (float)tg;
  r6[3] = tg ? glo : 0.0f;
  r6[4] = a_bi;
  r6[5] = a_con;

  // Deterministic wave32 tree reduction.
#pragma unroll
  for (int off = 16; off > 0; off >>= 1) {
#pragma unroll
    for (int j = 0; j < 6; ++j) r6[j] += __shfl_xor(r6[j], off, 32);
  }
  if (tx == 0) {
#pragma unroll
    for (int j = 0; j < 6; ++j) wred[ty][j] = r6[j];
  }
  __syncthreads();
  if (tid < 6) {
    float s = 0.0f;
#pragma unroll
    for (int wv = 0; wv < 8; ++wv) s += wred[wv][tid];  // fixed order
    const int blk = (blockIdx.z * gridDim.y + blockIdx.y) * gridDim.x + blockIdx.x;
    partial[blk * 6 + tid] = s;
  }
}

__global__ void __launch_bounds__(256)
bicon_finalize(const float* __restrict__ partial, float* __restrict__ out) {
  double acc[6] = {0, 0, 0, 0, 0, 0};
  for (int i = threadIdx.x; i < NBLK; i += 256) {
#pragma unroll
    for (int j = 0; j < 6; ++j) acc[j] += (double)partial[i * 6 + j];
  }
#pragma unroll
  for (int off = 16; off > 0; off >>= 1) {
#pragma unroll
    for (int j = 0; j < 6; ++j) acc[j] += __shfl_xor(acc[j], off, 32);
  }
  __shared__ double wp[8][6];
  const int lane = threadIdx.x & 31;
  const int wv = threadIdx.x >> 5;
  if (lane == 0) {
#pragma unroll
    for (int j = 0; j < 6; ++j) wp[wv][j] = acc[j];
  }
  __syncthreads();
  if (threadIdx.x == 0) {
    double s[6];
#pragma unroll
    for (int j = 0; j < 6; ++j) {
      double t = 0.0;
#pragma unroll
      for (int k = 0; k < 8; ++k) t += wp[k][j];  // fixed order
      s[j] = t;
    }
    const double NPIX = (double)BB * HH * WW;       // B*1*H*W
    const double NEL  = NPIX * CC;                  // B*C*H*W
    const double decouple = -s[0] / NPIX;
    const double gm = s[1], tg = s[2], gmt = s[3];
    const double cel = (gm + tg - 2.0 * gmt) / (gm + tg + 1e-6);
    const double bimap  = -s[4] / NEL;
    const double conmap = -s[5] / NEL;
    out[0] = (float)(0.8 * conmap + 0.2 * bimap + decouple + cel);
  }
}

extern "C" void kernel_launch(void* const* d_in, const int* in_sizes, int n_in,
                              void* d_out, int out_size, void* d_ws, size_t ws_size,
                              hipStream_t stream) {
  (void)in_sizes; (void)n_in; (void)out_size; (void)ws_size;
  const float* pred = (const float*)d_in[0];  // out: (16,8,352,352) f32
  const int*   tgt  = (const int*)d_in[1];    // target: (16,1,352,352) i32
  const int*   con  = (const int*)d_in[2];    // con_target: (16,8,352,352) i32
  float* partial = (float*)d_ws;              // NBLK*6 floats = ~186 KB

  dim3 grid(TILES_X, TILES_Y, BB);            // (11, 44, 16)
  bicon_main<<<grid, dim3(256), 0, stream>>>(pred, tgt, con, partial);
  bicon_finalize<<<1, dim3(256), 0, stream>>>(partial, (float*)d_out);
}